// XFeat_68350109549129
// MI455X (gfx1250) — compile-verified
//
#include <hip/hip_runtime.h>
#include <math.h>

typedef __attribute__((ext_vector_type(2))) float v2f;
typedef __attribute__((ext_vector_type(8))) float v8f;

#define HF   128
#define WFT  128
#define HH   1024
#define WWI  1024
#define TOPK 4096
#define CAP  131072  // per-batch candidate capacity (power of 2)

// ---------------------------------------------------------------------------
// Kernel 1: softmax over 65 channels + 8x8 pixel shuffle -> heat [B,1024,1024]
// ---------------------------------------------------------------------------
__global__ void k_heatmap(const float* __restrict__ K1, float* __restrict__ heat, int B) {
  int id = blockIdx.x * blockDim.x + threadIdx.x;
  int total = B * HF * WFT;
  if (id >= total) return;
  int wf = id % WFT;
  int hf = (id / WFT) % HF;
  int b  = id / (WFT * HF);
  const float* kp = K1 + (size_t)b * 65 * HF * WFT + (size_t)hf * WFT + wf;
  float m = -3.0e38f;
  for (int c = 0; c < 65; ++c) m = fmaxf(m, kp[(size_t)c * HF * WFT]);
  float s = 0.f;
  for (int c = 0; c < 65; ++c) s += expf(kp[(size_t)c * HF * WFT] - m);
  float inv = 1.0f / s;
  float* out = heat + (size_t)b * HH * WWI + (size_t)(hf * 8) * WWI + wf * 8;
  for (int i = 0; i < 8; ++i)
    for (int j = 0; j < 8; ++j)
      out[(size_t)i * WWI + j] = expf(kp[(size_t)(i * 8 + j) * HF * WFT] - m) * inv;
}

// ---------------------------------------------------------------------------
// Kernel 2: per-pixel 1/max(||M1||_c, eps) via V_WMMA_F32_16X16X4_F32 Gram.
// One wave32 = 16 pixels x 64 channels. A and B share registers (B = A^T for
// the f32 layout), 16 K=4 accumulation steps; Gram diagonal = sum of squares.
// ---------------------------------------------------------------------------
__global__ void k_wmma_norm(const float* __restrict__ M1, float* __restrict__ invmap) {
  int wave = threadIdx.x >> 5;
  int lane = threadIdx.x & 31;
  int T = blockIdx.x * (blockDim.x >> 5) + wave;  // tile id, tiles = B*1024 exactly
  int b   = T >> 10;
  int r   = T & 1023;
  int hf  = r >> 3;
  int wf0 = (r & 7) << 4;
  int m    = lane & 15;   // pixel within tile (M index)
  int half = lane >> 4;   // K half: lanes<16 -> K{0,1}, lanes>=16 -> K{2,3}
  const float* src = M1 + (size_t)b * 64 * HF * WFT + (size_t)hf * WFT + wf0 + m;
  v2f areg[16];
#pragma unroll
  for (int ch = 0; ch < 16; ++ch) {
    int c0 = 4 * ch + 2 * half;
    areg[ch].x = src[(size_t)c0 * HF * WFT];
    areg[ch].y = src[(size_t)(c0 + 1) * HF * WFT];
  }
  v8f acc = {0.f, 0.f, 0.f, 0.f, 0.f, 0.f, 0.f, 0.f};
#pragma unroll
  for (int ch = 0; ch < 16; ++ch) {
    acc = __builtin_amdgcn_wmma_f32_16x16x4_f32(false, areg[ch], false, areg[ch],
                                                (short)0, acc, false, false);
  }
  // Diagonal (M=N=p): p<8 -> lane p, vgpr p;  p>=8 -> lane p+16, vgpr p-8.
  int rr = lane & 7;
  float dg = acc[0];
  if (rr == 1) dg = acc[1];
  if (rr == 2) dg = acc[2];
  if (rr == 3) dg = acc[3];
  if (rr == 4) dg = acc[4];
  if (rr == 5) dg = acc[5];
  if (rr == 6) dg = acc[6];
  if (rr == 7) dg = acc[7];
  int srcLane = (m < 8) ? m : (m + 16);
  float nsq  = __shfl(dg, srcLane, 32);
  float invn = 1.0f / fmaxf(sqrtf(nsq), 1e-12f);
  if (lane == srcLane)
    invmap[(size_t)b * HF * WFT + (size_t)hf * WFT + wf0 + m] = invn;
}

// ---------------------------------------------------------------------------
// Kernel 3: init candidate keys to the pad key for score = -1.0 and zero counters
// ---------------------------------------------------------------------------
__global__ void k_init(unsigned long long* __restrict__ keys, int* __restrict__ cnt, int B) {
  int id = blockIdx.x * blockDim.x + threadIdx.x;
  if (id < B * CAP)
    keys[id] = (((unsigned long long)0x407FFFFFu) << 32) | 0xFFFFFFFFull; // orderable(-1.0), idx 0
  if (id < B) cnt[id] = 0;
}

// ---------------------------------------------------------------------------
// Kernel 4: 5x5 NMS + score; append survivors as sortable 64-bit keys
// key = orderable(score)<<32 | (0xFFFFFFFF - idx)  -> sort desc == (score desc, idx asc)
// ---------------------------------------------------------------------------
__global__ void k_score(const float* __restrict__ heat, const float* __restrict__ H1,
                        unsigned long long* __restrict__ keys, int* __restrict__ cnt, int B) {
  int id = blockIdx.x * blockDim.x + threadIdx.x;
  if (id >= B * HH * WWI) return;
  int x = id % WWI;
  int y = (id / WWI) % HH;
  int b = id / (WWI * HH);
  const float* hp = heat + (size_t)b * HH * WWI;
  float cv = hp[(size_t)y * WWI + x];
  float mx = cv;
  for (int dy = -2; dy <= 2; ++dy) {
    int yy = y + dy;
    if (yy < 0 || yy >= HH) continue;
    for (int dx = -2; dx <= 2; ++dx) {
      int xx = x + dx;
      if (xx < 0 || xx >= WWI) continue;
      mx = fmaxf(mx, hp[(size_t)yy * WWI + xx]);
    }
  }
  if (!(cv == mx && cv > 0.05f)) return;  // non-candidate -> stays as -1 pad
  if (x == 0 && y == 0) return;           // sc[:,0] = -1
  float fx = (float)x, fy = (float)y;
  // nearest sample of heat (grid scale 1024/1023, round-half-even, zero pad)
  float cxn = fx * (float)(1024.0 / 1023.0) - 0.5f;
  float cyn = fy * (float)(1024.0 / 1023.0) - 0.5f;
  int xi = (int)rintf(cxn);
  int yi = (int)rintf(cyn);
  float sn = (xi >= 0 && xi < WWI && yi >= 0 && yi < HH) ? hp[(size_t)yi * WWI + xi] : 0.f;
  // bilinear sample of H1 (128x128, scale 128/1023, zero pad)
  float cxb = fx * (float)(128.0 / 1023.0) - 0.5f;
  float cyb = fy * (float)(128.0 / 1023.0) - 0.5f;
  float x0f = floorf(cxb), y0f = floorf(cyb);
  float wx = cxb - x0f, wy = cyb - y0f;
  int x0 = (int)x0f, y0 = (int)y0f;
  const float* h1 = H1 + (size_t)b * HF * WFT;
  float v00 = (x0 >= 0     && x0 < WFT     && y0 >= 0     && y0 < HF)     ? h1[(size_t)y0 * WFT + x0]           : 0.f;
  float v01 = (x0 + 1 >= 0 && x0 + 1 < WFT && y0 >= 0     && y0 < HF)     ? h1[(size_t)y0 * WFT + x0 + 1]       : 0.f;
  float v10 = (x0 >= 0     && x0 < WFT     && y0 + 1 >= 0 && y0 + 1 < HF) ? h1[(size_t)(y0 + 1) * WFT + x0]     : 0.f;
  float v11 = (x0 + 1 >= 0 && x0 + 1 < WFT && y0 + 1 >= 0 && y0 + 1 < HF) ? h1[(size_t)(y0 + 1) * WFT + x0 + 1] : 0.f;
  float sb = v00 * (1.f - wy) * (1.f - wx) + v01 * (1.f - wy) * wx
           + v10 * wy * (1.f - wx)         + v11 * wy * wx;
  float sc = sn * sb;  // >= 0 always (softmax * uniform, zero-padded)
  unsigned idx = (unsigned)(y * WWI + x);
  unsigned u = __float_as_uint(sc);
  u = (u & 0x80000000u) ? ~u : (u | 0x80000000u);
  unsigned long long key =
      (((unsigned long long)u) << 32) | (unsigned long long)(0xFFFFFFFFu - idx);
  int slot = atomicAdd(&cnt[b], 1);
  if (slot < CAP) keys[(size_t)b * CAP + slot] = key;
}

// ---------------------------------------------------------------------------
// Kernel 5: one compare-exchange step of segmented bitonic sort (descending)
// ---------------------------------------------------------------------------
__global__ void k_bitonic(unsigned long long* __restrict__ keys, int k, int j, int B) {
  int t = blockIdx.x * blockDim.x + threadIdx.x;
  if (t >= B * CAP) return;
  int seg = t / CAP;
  int li  = t & (CAP - 1);
  int pj = li ^ j;
  if (pj <= li) return;
  size_t i1 = (size_t)seg * CAP + li;
  size_t i2 = (size_t)seg * CAP + pj;
  unsigned long long a = keys[i1], c = keys[i2];
  bool up = (li & k) == 0;                 // descending region for final desc order
  bool sw = up ? (a < c) : (a > c);
  if (sw) { keys[i1] = c; keys[i2] = a; }
}

// ---------------------------------------------------------------------------
// Kernel 6: decode top-K -> mk [B,K,2] and vals [B,K]
// ---------------------------------------------------------------------------
__global__ void k_emit(const unsigned long long* __restrict__ keys, float* __restrict__ out, int B) {
  int id = blockIdx.x * blockDim.x + threadIdx.x;
  if (id >= B * TOPK) return;
  int k = id % TOPK;
  int b = id / TOPK;
  unsigned long long key = keys[(size_t)b * CAP + k];
  unsigned u = (unsigned)(key >> 32);
  float val = (u & 0x80000000u) ? __uint_as_float(u & 0x7FFFFFFFu) : __uint_as_float(~u);
  unsigned idx = 0xFFFFFFFFu - (unsigned)(key & 0xFFFFFFFFull);
  float mkx = (float)(idx % WWI);
  float mky = (float)(idx / WWI);
  if (!(val > 0.f)) { mkx = 0.f; mky = 0.f; }   // torch-style (0,0) padding
  out[(size_t)id * 2 + 0] = mkx;
  out[(size_t)id * 2 + 1] = mky;
  out[(size_t)B * TOPK * 2 + id] = val;
}

// ---------------------------------------------------------------------------
// Kernel 7: bilinear-sample normalized descriptors at mk, L2-normalize.
// block = 64 threads = 64 channels of one keypoint.
// ---------------------------------------------------------------------------
__global__ void k_feats(const float* __restrict__ M1, const float* __restrict__ invmap,
                        float* __restrict__ out, int B) {
  int kp = blockIdx.x;      // [0, B*TOPK)
  int c  = threadIdx.x;     // channel
  int b  = kp / TOPK;
  float mkx = out[(size_t)kp * 2 + 0];
  float mky = out[(size_t)kp * 2 + 1];
  float cx = mkx * (float)(128.0 / 1023.0) - 0.5f;
  float cy = mky * (float)(128.0 / 1023.0) - 0.5f;
  float x0f = floorf(cx), y0f = floorf(cy);
  float wx = cx - x0f, wy = cy - y0f;
  int x0 = (int)x0f, y0 = (int)y0f;
  const float* mb = M1 + ((size_t)b * 64 + c) * HF * WFT;
  const float* im = invmap + (size_t)b * HF * WFT;
  float v = 0.f;
  {
    int yy = y0, xx = x0;
    if (xx >= 0 && xx < WFT && yy >= 0 && yy < HF)
      v += mb[(size_t)yy * WFT + xx] * im[(size_t)yy * WFT + xx] * (1.f - wy) * (1.f - wx);
  }
  {
    int yy = y0, xx = x0 + 1;
    if (xx >= 0 && xx < WFT && yy >= 0 && yy < HF)
      v += mb[(size_t)yy * WFT + xx] * im[(size_t)yy * WFT + xx] * (1.f - wy) * wx;
  }
  {
    int yy = y0 + 1, xx = x0;
    if (xx >= 0 && xx < WFT && yy >= 0 && yy < HF)
      v += mb[(size_t)yy * WFT + xx] * im[(size_t)yy * WFT + xx] * wy * (1.f - wx);
  }
  {
    int yy = y0 + 1, xx = x0 + 1;
    if (xx >= 0 && xx < WFT && yy >= 0 && yy < HF)
      v += mb[(size_t)yy * WFT + xx] * im[(size_t)yy * WFT + xx] * wy * wx;
  }
  __shared__ float red[64];
  red[c] = v * v;
  __syncthreads();
  for (int s = 32; s > 0; s >>= 1) {
    if (c < s) red[c] += red[c + s];
    __syncthreads();
  }
  float denom = fmaxf(sqrtf(red[0]), 1e-12f);
  out[(size_t)B * TOPK * 3 + (size_t)kp * 64 + c] = v / denom;
}

// ---------------------------------------------------------------------------
extern "C" void kernel_launch(void* const* d_in, const int* in_sizes, int n_in,
                              void* d_out, int out_size, void* d_ws, size_t ws_size,
                              hipStream_t stream) {
  (void)n_in; (void)out_size; (void)ws_size;
  const float* M1 = (const float*)d_in[0];  // [B,64,128,128]
  const float* K1 = (const float*)d_in[1];  // [B,65,128,128]
  const float* H1 = (const float*)d_in[2];  // [B,1,128,128]
  int B = in_sizes[2] / (HF * WFT);
  float* out = (float*)d_out;

  // workspace layout (8-byte aligned throughout)
  float* heat   = (float*)d_ws;                              // B*1024*1024
  float* invmap = heat + (size_t)B * HH * WWI;               // B*128*128
  unsigned long long* keys =
      (unsigned long long*)(invmap + (size_t)B * HF * WFT);  // B*CAP
  int* cnt = (int*)(keys + (size_t)B * CAP);                 // B

  k_heatmap<<<(B * HF * WFT + 255) / 256, 256, 0, stream>>>(K1, heat, B);
  k_wmma_norm<<<B * 128, 256, 0, stream>>>(M1, invmap);      // B*1024 tiles, 8 waves/block
  k_init<<<(B * CAP + 255) / 256, 256, 0, stream>>>(keys, cnt, B);
  k_score<<<(B * HH * WWI + 255) / 256, 256, 0, stream>>>(heat, H1, keys, cnt, B);
  for (int k = 2; k <= CAP; k <<= 1)
    for (int j = k >> 1; j >= 1; j >>= 1)
      k_bitonic<<<(B * CAP + 255) / 256, 256, 0, stream>>>(keys, k, j, B);
  k_emit<<<(B * TOPK + 255) / 256, 256, 0, stream>>>(keys, out, B);
  k_feats<<<B * TOPK, 64, 0, stream>>>(M1, invmap, out, B);
}